// Attn_loc_distance_71090298683716
// MI455X (gfx1250) — compile-verified
//
#include <hip/hip_runtime.h>

// Attn_loc_distance: out[r, :] = 1 / where(D[venueid2coor[inputs_poi[r]], :] == 0, BIG, ...)
// Pure streaming gather: bandwidth-bound (~512 MB HBM traffic, floor ~22us @ 23.3 TB/s).
// Strategy: 1 workgroup (8 wave32) per row, b128 loads, NT b128 stores, L2 prefetch.

#define BIG_DIST 9999999.99f

// Clang-native 4-wide float vector: accepted by __builtin_nontemporal_store,
// same 16B layout as float4, lowers to b128 vector memory ops.
typedef __attribute__((ext_vector_type(4))) float v4f;

__global__ __launch_bounds__(256) void attn_loc_distance_kernel(
    const int*   __restrict__ venueid2coor,   // [V]
    const int*   __restrict__ inputs_poi,     // [B*S]
    const float* __restrict__ dist,           // [N, N]
    float*       __restrict__ out,            // [B*S, N]
    int n_cols)                               // N (= N_POI)
{
    const int row = blockIdx.x;               // which (batch, step) row
    // Wave-uniform index chain: lowers to scalar (s_load) fetches.
    const int       poi     = inputs_poi[row];
    const long long src_row = (long long)venueid2coor[poi];

    const v4f* __restrict__ src = (const v4f*)(dist + src_row * (long long)n_cols);
    v4f*       __restrict__ dst = (v4f*)(out + (long long)row * (long long)n_cols);

    const int nvec = n_cols >> 2;             // 2500 float4 per row for N=10000

    for (int i = threadIdx.x; i < nvec; i += blockDim.x) {
        // Speculative L2 prefetch two block-strides ahead on this row's stream
        // (gfx1250 global_prefetch_b8; OOB speculative prefetches are dropped).
        __builtin_prefetch((const void*)(src + i + 2 * blockDim.x), 0, 1);

        v4f v = src[i];                       // global_load_b128 (RT: keep rows in L2)
        v4f r;
        r.x = 1.0f / ((v.x == 0.0f) ? BIG_DIST : v.x);
        r.y = 1.0f / ((v.y == 0.0f) ? BIG_DIST : v.y);
        r.z = 1.0f / ((v.z == 0.0f) ? BIG_DIST : v.z);
        r.w = 1.0f / ((v.w == 0.0f) ? BIG_DIST : v.w);

        // Output is write-once/never-read: non-temporal b128 store so the 256MB
        // output stream does not evict the reusable distance rows from L2.
        __builtin_nontemporal_store(r, dst + i);
    }

    // Scalar tail (not taken for N=10000, kept for generality).
    const int tail_start = nvec << 2;
    const float* __restrict__ srcs = (const float*)src;
    float*       __restrict__ dsts = (float*)dst;
    for (int i = tail_start + threadIdx.x; i < n_cols; i += blockDim.x) {
        float v = srcs[i];
        float m = (v == 0.0f) ? BIG_DIST : v;
        __builtin_nontemporal_store(1.0f / m, dsts + i);
    }
}

extern "C" void kernel_launch(void* const* d_in, const int* in_sizes, int n_in,
                              void* d_out, int out_size, void* d_ws, size_t ws_size,
                              hipStream_t stream) {
    // setup_inputs() order: venueid2coor [V], inputs_poi [B*S], poi_distance_matrix [N*N]
    const int*   venueid2coor = (const int*)d_in[0];
    const int*   inputs_poi   = (const int*)d_in[1];
    const float* dist         = (const float*)d_in[2];
    float*       out          = (float*)d_out;

    const int rows   = in_sizes[1];            // B*S = 6400
    const int n_cols = out_size / rows;        // N_POI = 10000

    dim3 grid((unsigned)rows);
    dim3 block(256);                           // 8 wave32 per WGP slot
    attn_loc_distance_kernel<<<grid, block, 0, stream>>>(
        venueid2coor, inputs_poi, dist, out, n_cols);
}